// SimpleTP_20272245637494
// MI455X (gfx1250) — compile-verified
//
#include <hip/hip_runtime.h>

// out[k] = sum_n sum_ij frac[n,i] frac[n,j] cg[i,j,k]
//        = contract( M[i,j] = sum_n frac[n,i] frac[n,j],  cg )
//
// Memory-bound: 96 MiB streamed once -> ~4.3 us floor at 23.3 TB/s.
// Kernel 1 streams frac with nontemporal b128 loads and reduces the 3x3
// second-moment (6 unique values) per block. Kernel 2 (one wave32) sums the
// block partials and does the 9x9 CG contraction with v_wmma_f32_16x16x4_f32.

typedef float f4  __attribute__((ext_vector_type(4)));
typedef float v2f __attribute__((ext_vector_type(2)));
typedef float v8f __attribute__((ext_vector_type(8)));

#define ACCROW(X, Y, Z)                                                        \
  do {                                                                         \
    xx = __builtin_fmaf((X), (X), xx);                                         \
    xy = __builtin_fmaf((X), (Y), xy);                                         \
    xz = __builtin_fmaf((X), (Z), xz);                                         \
    yy = __builtin_fmaf((Y), (Y), yy);                                         \
    yz = __builtin_fmaf((Y), (Z), yz);                                         \
    zz = __builtin_fmaf((Z), (Z), zz);                                         \
  } while (0)

__global__ __launch_bounds__(256) void tp_moment_kernel(
    const float* __restrict__ frac, float* __restrict__ partials, int nQuads,
    int nRows) {
  const f4* fq = (const f4*)frac;
  const int tid = blockIdx.x * blockDim.x + threadIdx.x;
  const int stride = gridDim.x * blockDim.x;

  float xx = 0.f, xy = 0.f, xz = 0.f, yy = 0.f, yz = 0.f, zz = 0.f;

  // 4 rows (48 B) per lane per iteration: three NT 128-bit loads.
  for (int q = tid; q < nQuads; q += stride) {
    f4 f0 = __builtin_nontemporal_load(fq + 3 * q + 0);
    f4 f1 = __builtin_nontemporal_load(fq + 3 * q + 1);
    f4 f2 = __builtin_nontemporal_load(fq + 3 * q + 2);
    ACCROW(f0.x, f0.y, f0.z);  // row 4q+0
    ACCROW(f0.w, f1.x, f1.y);  // row 4q+1
    ACCROW(f1.z, f1.w, f2.x);  // row 4q+2
    ACCROW(f2.y, f2.z, f2.w);  // row 4q+3
  }
  // Tail rows (none for N = 2^23, kept for generality).
  for (int r = 4 * nQuads + tid; r < nRows; r += stride) {
    float x = frac[3 * r + 0], y = frac[3 * r + 1], z = frac[3 * r + 2];
    ACCROW(x, y, z);
  }

  float acc[6] = {xx, xy, xz, yy, yz, zz};
#pragma unroll
  for (int i = 0; i < 6; ++i)
    for (int off = 16; off; off >>= 1) acc[i] += __shfl_xor(acc[i], off, 32);

  __shared__ float red[8][6];
  const int lane = threadIdx.x & 31;
  const int wid = threadIdx.x >> 5;
  if (lane == 0) {
#pragma unroll
    for (int i = 0; i < 6; ++i) red[wid][i] = acc[i];
  }
  __syncthreads();
  if (threadIdx.x < 6) {
    float s = 0.f;
    const int nw = blockDim.x >> 5;
    for (int w = 0; w < nw; ++w) s += red[w][threadIdx.x];
    partials[blockIdx.x * 6 + threadIdx.x] = s;
  }
}

// One wave32. Sums per-block partials, then out[k] = sum_p Mflat[p]*CG[p,k]
// via three chained v_wmma_f32_16x16x4_f32 (K = 0..3, 4..7, 8 w/ zero pad).
__global__ __launch_bounds__(32) void tp_finalize_kernel(
    const float* __restrict__ partials, int nPart, const float* __restrict__ cg,
    float* __restrict__ out) {
  const int lane = threadIdx.x;  // 0..31
  float s[6] = {0.f, 0.f, 0.f, 0.f, 0.f, 0.f};
  for (int b = lane; b < nPart; b += 32) {
#pragma unroll
    for (int i = 0; i < 6; ++i) s[i] += partials[b * 6 + i];
  }
#pragma unroll
  for (int i = 0; i < 6; ++i)
    for (int off = 16; off; off >>= 1) s[i] += __shfl_xor(s[i], off, 32);

  // Mflat[p], p = i*3+j, symmetric 3x3 second moment. All lanes hold it.
  const float M[9] = {s[0], s[1], s[2], s[1], s[3], s[4], s[2], s[4], s[5]};

#if __has_builtin(__builtin_amdgcn_wmma_f32_16x16x4_f32)
  // A: 16x4 f32, only row M=0 used. ISA layout: VGPR0 = K0 (lanes 0-15) /
  // K2 (lanes 16-31); VGPR1 = K1 / K3. B: 4x16 f32, column N = lane%16,
  // VGPR0 = K0/K2, VGPR1 = K1/K3. D row M=0 -> VGPR0, lanes 0-15.
  const int n = lane & 15;
  const int half = lane >> 4;
  v8f c = {0.f, 0.f, 0.f, 0.f, 0.f, 0.f, 0.f, 0.f};
#pragma unroll
  for (int cc = 0; cc < 3; ++cc) {
    const int kb = 4 * cc + 2 * half;  // K index held by VGPR0 on this lane
    v2f a, b;
    a.x = (n == 0 && kb + 0 < 9) ? M[kb + 0] : 0.f;
    a.y = (n == 0 && kb + 1 < 9) ? M[kb + 1] : 0.f;
    b.x = (kb + 0 < 9 && n < 9) ? cg[(kb + 0) * 9 + n] : 0.f;
    b.y = (kb + 1 < 9 && n < 9) ? cg[(kb + 1) * 9 + n] : 0.f;
    c = __builtin_amdgcn_wmma_f32_16x16x4_f32(
        /*neg_a=*/false, a, /*neg_b=*/false, b,
        /*c_mod=*/(short)0, c, /*reuse_a=*/false, /*reuse_b=*/false);
  }
  if (lane < 9) out[lane] = c[0];
#else
  // Exact scalar fallback (host pass / builtin missing).
  if (lane < 9) {
    float o = 0.f;
#pragma unroll
    for (int p = 0; p < 9; ++p) o += M[p] * cg[p * 9 + lane];
    out[lane] = o;
  }
#endif
}

extern "C" void kernel_launch(void* const* d_in, const int* in_sizes, int n_in,
                              void* d_out, int out_size, void* d_ws,
                              size_t ws_size, hipStream_t stream) {
  const float* frac = (const float*)d_in[0];
  const float* cg = (const float*)d_in[1];  // [3,3,9] flat, 81 floats
  float* out = (float*)d_out;
  float* ws = (float*)d_ws;

  const long long nElem = (long long)in_sizes[0];  // 3*N
  const int nRows = (int)(nElem / 3);
  const int nQuads = nRows / 4;

  const int threads = 256;
  int blocks = 1024;  // ~262k threads: 8 quads (384 B) each at N = 2^23
  const long long maxBlocksWs = (long long)(ws_size / (6 * sizeof(float)));
  if ((long long)blocks > maxBlocksWs) blocks = (int)maxBlocksWs;
  if (blocks < 1) blocks = 1;

  tp_moment_kernel<<<blocks, threads, 0, stream>>>(frac, ws, nQuads, nRows);
  tp_finalize_kernel<<<1, 32, 0, stream>>>(ws, blocks, cg, out);
}